// DisulfideWholePoseScoringModule_56530359550769
// MI455X (gfx1250) — compile-verified
//
#include <hip/hip_runtime.h>
#include <cstdint>
#include <cstddef>

// ---------------------------------------------------------------------------
// Disulfide whole-pose scoring for MI455X (gfx1250, wave32).
// P=512 poses, B=1024 blocks, APB=16 atoms/block. Output: 512 floats.
// Latency/gather-bound (~35 MB touched -> ~1.5us at 23.3 TB/s); no matmul
// structure, so no WMMA. CDNA5 path used: TDM async tensor_load_to_lds to
// stage the per-pose index rows (reused 2x) into LDS + s_wait_tensorcnt,
// wave32 shuffle reduction.
// ---------------------------------------------------------------------------

namespace {

constexpr int kP       = 512;
constexpr int kB       = 1024;
constexpr int kAPB     = 16;
constexpr int kNConn   = 3;
constexpr int kThreads = 256;   // 8 wave32

#define LOG_SQRT_2PI_F 0.9189385332046727f

struct V3 { float x, y, z; };
__device__ __forceinline__ V3   v3sub(V3 a, V3 b) { return {a.x - b.x, a.y - b.y, a.z - b.z}; }
__device__ __forceinline__ float v3dot(V3 a, V3 b) { return a.x * b.x + a.y * b.y + a.z * b.z; }
__device__ __forceinline__ V3   v3cross(V3 a, V3 b) {
  return {a.y * b.z - a.z * b.y, a.z * b.x - a.x * b.z, a.x * b.y - a.y * b.x};
}
__device__ __forceinline__ V3   v3scale(V3 a, float s) { return {a.x * s, a.y * s, a.z * s}; }

__device__ __forceinline__ float angle3(V3 a, V3 b, V3 c) {
  V3 u = v3sub(a, b), v = v3sub(c, b);
  float ca = v3dot(u, v) * rsqrtf(v3dot(u, u) * v3dot(v, v) + 1e-12f);
  ca = fminf(fmaxf(ca, -1.0f + 1e-7f), 1.0f - 1e-7f);
  return acosf(ca);
}

__device__ __forceinline__ float dihedral4(V3 p0, V3 p1, V3 p2, V3 p3) {
  V3 b1 = v3sub(p1, p0), b2 = v3sub(p2, p1), b3 = v3sub(p3, p2);
  V3 n1 = v3cross(b1, b2), n2 = v3cross(b2, b3);
  V3 b2n = v3scale(b2, rsqrtf(v3dot(b2, b2) + 1e-12f));
  V3 m1 = v3cross(n1, b2n);
  return atan2f(v3dot(m1, n2), v3dot(n1, n2));
}

// log(Phi(x)) valid over the full range needed here (|x| up to ~3e2).
__device__ __forceinline__ float log_ndtr(float x) {
  if (x > -2.0f) {
    return logf(0.5f * erfcf(-x * 0.70710678118654752f));
  }
  float ix2  = 1.0f / (x * x);
  float corr = log1pf(ix2 * (-1.0f + ix2 * (3.0f - 15.0f * ix2)));
  return -0.5f * x * x - LOG_SQRT_2PI_F - logf(-x) + corr;
}

__device__ __forceinline__ float nlse2(float c0, float c1) {
  float m = fmaxf(c0, c1);
  return -(m + logf(expf(c0 - m) + expf(c1 - m)));
}
__device__ __forceinline__ float nlse3(float c0, float c1, float c2) {
  float m = fmaxf(fmaxf(c0, c1), c2);
  return -(m + logf(expf(c0 - m) + expf(c1 - m) + expf(c2 - m)));
}

} // namespace

// ---------------------------------------------------------------------------
// TDM staging. This toolchain exposes the 6-arg clang-23 builtin form:
//   (uint32x4 g0, int32x8 g1, int32x4 g2, int32x4 g3, int32x8 g4, i32 cpol)
// Falls back to plain coalesced loads if the builtin is unavailable.
// ---------------------------------------------------------------------------
#if defined(__HIP_DEVICE_COMPILE__) && defined(__gfx1250__) && \
    __has_builtin(__builtin_amdgcn_tensor_load_to_lds)
#define USE_TDM 1

typedef __attribute__((ext_vector_type(4))) unsigned int tdm_u32x4;
typedef __attribute__((ext_vector_type(8))) int          tdm_i32x8;
typedef __attribute__((ext_vector_type(4))) int          tdm_i32x4;

// 1-D tile load: ndwords DWORDs from gsrc -> LDS byte offset lds_byte_off.
__device__ __forceinline__ void tdm_load_dwords(const void* gsrc,
                                                unsigned lds_byte_off,
                                                unsigned ndwords) {
  unsigned long long ga = (unsigned long long)(uintptr_t)gsrc;
  tdm_u32x4 g0;
  g0[0] = 1u;                                            // count=1, user descriptor
  g0[1] = lds_byte_off;                                  // lds_addr
  g0[2] = (unsigned)(ga & 0xFFFFFFFFu);                  // global_addr[31:0]
  g0[3] = (unsigned)((ga >> 32) & 0x01FFFFFFu) |         // global_addr[56:32]
          (2u << 30);                                    // type=2 ("image")
  tdm_i32x8 g1;
  g1[0] = (int)(2u << 16);                               // data_size=2 -> 4 bytes
  g1[1] = (int)((ndwords & 0xFFFFu) << 16);              // tensor_dim0[15:0]
  g1[2] = (int)(((ndwords >> 16) & 0xFFFFu) |            // tensor_dim0[31:16]
                (1u << 16));                             // tensor_dim1 = 1
  g1[3] = (int)((ndwords & 0xFFFFu) << 16);              // tile_dim0
  g1[4] = 1;                                             // tile_dim1 = 1
  g1[5] = (int)ndwords;                                  // tensor_dim0_stride lo32
  g1[6] = 0;                                             // stride hi / dim1_stride
  g1[7] = 0;
  tdm_i32x4 z4 = {0, 0, 0, 0};
  tdm_i32x8 z8 = {0, 0, 0, 0, 0, 0, 0, 0};
  __builtin_amdgcn_tensor_load_to_lds(g0, g1, z4, z4, z8, 0);
}
#else
#define USE_TDM 0
#endif

__global__ __launch_bounds__(kThreads) void dslf_score_kernel(
    const float* __restrict__ coords,     // (P, B*APB, 3)
    const int*   __restrict__ blk_off,    // (P, B)
    const int*   __restrict__ blk_type,   // (P, B)
    const int*   __restrict__ ibc,        // (P, B, NCONN, 2)
    const int*   __restrict__ bt_dslf,    // (2,)
    const int*   __restrict__ adoc,       // (2, NCONN, APB)
    const float* __restrict__ gp,         // (21,)
    float*       __restrict__ out)        // (P,)
{
  __shared__ int   s_off[kB];
  __shared__ int   s_bt[kB];
  __shared__ float s_red[kThreads / 32];

  const int p = blockIdx.x;
  const int t = threadIdx.x;

#if USE_TDM
  if (t < 32) {  // wave 0 issues the async tensor DMA for the whole workgroup
    tdm_load_dwords(blk_off  + (size_t)p * kB,
                    (unsigned)(uintptr_t)(void*)s_off, (unsigned)kB);
    tdm_load_dwords(blk_type + (size_t)p * kB,
                    (unsigned)(uintptr_t)(void*)s_bt, (unsigned)kB);
    __builtin_amdgcn_s_wait_tensorcnt(0);
  }
  __syncthreads();
#else
  for (int i = t; i < kB; i += kThreads) {
    s_off[i] = blk_off[(size_t)p * kB + i];
    s_bt[i]  = blk_type[(size_t)p * kB + i];
  }
  __syncthreads();
#endif

  // Global params (uniform scalar loads).
  const float d_loc = gp[0], d_scale = gp[1], d_shape = gp[2];
  const float a_logA = gp[3], a_kappa = gp[4], a_mu = gp[5];
  const float ss_lA0 = gp[6],  ss_k0 = gp[7],  ss_mu0 = gp[8];
  const float ss_lA1 = gp[9],  ss_k1 = gp[10], ss_mu1 = gp[11];
  const float cs_lA0 = gp[12], cs_mu0 = gp[13], cs_k0 = gp[14];
  const float cs_lA1 = gp[15], cs_mu1 = gp[16], cs_k1 = gp[17];
  const float cs_lA2 = gp[18], cs_mu2 = gp[19], cs_k2 = gp[20];
  const float log_2_over_scale = logf(2.0f / d_scale);

  const float* cpose = coords + (size_t)p * (size_t)(kB * kAPB * 3);

  float acc = 0.0f;
  for (int b = t; b < kB; b += kThreads) {
    const int bt    = s_bt[b];
    const int dconn = bt_dslf[bt];
    const int dc    = dconn < 0 ? 0 : dconn;
    const size_t nidx = (((size_t)p * kB + (size_t)b) * kNConn + (size_t)dc) * 2;
    const int ob = ibc[nidx + 0];
    const int oc = ibc[nidx + 1];
    if (dconn < 0 || ob <= b) continue;   // mask: (dconn>=0) & (ob > b)

    const int oc_s = oc < 0 ? 0 : oc;
    const int obt  = s_bt[ob];
    const int off1 = s_off[b];
    const int off2 = s_off[ob];
    const int* down1 = adoc + ((size_t)bt  * kNConn + (size_t)dc)   * kAPB;
    const int* down2 = adoc + ((size_t)obt * kNConn + (size_t)oc_s) * kAPB;

    auto atom = [&](int off, int dn) -> V3 {
      const float* q = cpose + (size_t)(off + dn) * 3;
      return V3{q[0], q[1], q[2]};
    };
    const V3 sg1 = atom(off1, down1[0]);
    const V3 cb1 = atom(off1, down1[1]);
    const V3 ca1 = atom(off1, down1[2]);
    const V3 sg2 = atom(off2, down2[0]);
    const V3 cb2 = atom(off2, down2[1]);
    const V3 ca2 = atom(off2, down2[2]);

    // Distance score (skew-normal log-pdf).
    const V3 dv = v3sub(sg1, sg2);
    const float d = sqrtf(v3dot(dv, dv) + 1e-12f);
    const float z = (d - d_loc) / d_scale;
    const float score_d =
        -(log_2_over_scale - 0.5f * z * z - LOG_SQRT_2PI_F + log_ndtr(d_shape * z));

    // Angle scores.
    const float ang1 = angle3(cb1, sg1, sg2);
    const float ang2 = angle3(sg1, sg2, cb2);
    const float score_a = -(a_logA + a_kappa * cosf(ang1 - a_mu))
                          -(a_logA + a_kappa * cosf(ang2 - a_mu));

    // SS dihedral: 2-component von Mises mixture.
    const float xss = dihedral4(cb1, sg1, sg2, cb2);
    const float score_ss = nlse2(ss_lA0 + ss_k0 * cosf(xss - ss_mu0),
                                 ss_lA1 + ss_k1 * cosf(xss - ss_mu1));

    // CS dihedrals: 3-component von Mises mixtures.
    const float x1 = dihedral4(ca1, cb1, sg1, sg2);
    const float x2 = dihedral4(sg1, sg2, cb2, ca2);
    const float score_cs = nlse3(cs_lA0 + cs_k0 * cosf(x1 - cs_mu0),
                                 cs_lA1 + cs_k1 * cosf(x1 - cs_mu1),
                                 cs_lA2 + cs_k2 * cosf(x1 - cs_mu2))
                         + nlse3(cs_lA0 + cs_k0 * cosf(x2 - cs_mu0),
                                 cs_lA1 + cs_k1 * cosf(x2 - cs_mu1),
                                 cs_lA2 + cs_k2 * cosf(x2 - cs_mu2));

    acc += score_d + score_a + score_ss + score_cs;
  }

  // wave32 shuffle reduction, then cross-wave via LDS. Deterministic order.
  for (int o = 16; o > 0; o >>= 1) acc += __shfl_down(acc, o, 32);
  if ((t & 31) == 0) s_red[t >> 5] = acc;
  __syncthreads();
  if (t == 0) {
    float s = 0.0f;
#pragma unroll
    for (int w = 0; w < kThreads / 32; ++w) s += s_red[w];
    out[p] = s;
  }
}

extern "C" void kernel_launch(void* const* d_in, const int* in_sizes, int n_in,
                              void* d_out, int out_size, void* d_ws, size_t ws_size,
                              hipStream_t stream) {
  (void)in_sizes; (void)n_in; (void)out_size; (void)d_ws; (void)ws_size;
  const float* coords   = (const float*)d_in[0];
  const int*   blk_off  = (const int*)d_in[1];
  const int*   blk_type = (const int*)d_in[2];
  const int*   ibc      = (const int*)d_in[3];
  const int*   bt_dslf  = (const int*)d_in[4];
  const int*   adoc     = (const int*)d_in[5];
  const float* gp       = (const float*)d_in[6];
  float*       out      = (float*)d_out;

  dslf_score_kernel<<<kP, kThreads, 0, stream>>>(
      coords, blk_off, blk_type, ibc, bt_dslf, adoc, gp, out);
}